// QuerySpecificClusteringModel_3367254360580
// MI455X (gfx1250) — compile-verified
//
#include <hip/hip_runtime.h>
#include <hip/hip_bf16.h>

typedef __attribute__((ext_vector_type(2))) float v2f;
typedef __attribute__((ext_vector_type(8))) float v8f;

#define N_ROWS 16384
#define DDIM   256
#define KCL    64
#define RPB    64            // rows of X per block
#define NBLK   (N_ROWS / RPB)
#define XPAD   260           // LDS row stride for X/C tiles (bank-conflict pad)
#define PPAD   68            // LDS row stride for P / a tile
#define SMEM_FLOATS (2 * RPB * XPAD + RPB * PPAD + KCL)
#define SMEM_BYTES  (SMEM_FLOATS * sizeof(float))
#define EPS_   1e-6f

__device__ __forceinline__ v8f wmma_f32(v2f a, v2f b, v8f c) {
  // D = A(16x4 f32) * B(4x16 f32) + C(16x16 f32)  -> v_wmma_f32_16x16x4_f32
  return __builtin_amdgcn_wmma_f32_16x16x4_f32(false, a, false, b, (short)0, c,
                                               false, false);
}

// Async direct global->LDS copy (GLOBAL_LOAD_ASYNC_TO_LDS_B128, ASYNCcnt).
// lds_addr: per-lane LDS byte address (low 32 bits of generic shared pointer);
// goff: per-lane byte offset from sbase (GVS addressing mode).
__device__ __forceinline__ void async_copy_b128(unsigned int lds_addr,
                                                unsigned int goff,
                                                unsigned long long sbase) {
  asm volatile("global_load_async_to_lds_b128 %0, %1, %2 offset:0"
               :
               : "v"(lds_addr), "v"(goff), "s"(sbase)
               : "memory");
}
__device__ __forceinline__ void wait_asynccnt0() {
  asm volatile("s_wait_asynccnt 0x0" ::: "memory");
}

// ---------------------------------------------------------------------------
// x2[r] = sum_d X[r][d]^2
__global__ __launch_bounds__(256) void x2_kernel(const float* __restrict__ X,
                                                 float* __restrict__ x2) {
  const int r = blockIdx.x * 256 + threadIdx.x;
  const float4* p = (const float4*)(X + (size_t)r * DDIM);
  float s = 0.f;
  for (int j = 0; j < DDIM / 4; ++j) {
    float4 v = p[j];
    s += v.x * v.x + v.y * v.y + v.z * v.z + v.w * v.w;
  }
  x2[r] = s;
}

// ---------------------------------------------------------------------------
// C <- C_init ; c2[k] = ||C[k]||^2 ; S <- 0 ; cnt <- 0
__global__ __launch_bounds__(256) void initC_kernel(const float* __restrict__ Cin,
                                                    float* __restrict__ C,
                                                    float* __restrict__ c2,
                                                    float* __restrict__ S,
                                                    float* __restrict__ cnt) {
  const int k = blockIdx.x;
  const int d = threadIdx.x;
  const float v = Cin[(size_t)k * DDIM + d];
  C[(size_t)k * DDIM + d] = v;
  S[(size_t)k * DDIM + d] = 0.f;
  __shared__ float red[256];
  red[d] = v * v;
  __syncthreads();
  for (int s = 128; s > 0; s >>= 1) {
    if (d < s) red[d] += red[d + s];
    __syncthreads();
  }
  if (d == 0) { c2[k] = red[0]; cnt[k] = 0.f; }
}

// ---------------------------------------------------------------------------
// C <- S / (cnt + eps) ; c2 <- row norms ; zero S, cnt for next iteration
__global__ __launch_bounds__(256) void update_kernel(float* __restrict__ C,
                                                     float* __restrict__ c2,
                                                     float* __restrict__ S,
                                                     float* __restrict__ cnt) {
  const int k = blockIdx.x;
  const int d = threadIdx.x;
  const float cn = cnt[k] + EPS_;
  const float v = S[(size_t)k * DDIM + d] / cn;
  C[(size_t)k * DDIM + d] = v;
  S[(size_t)k * DDIM + d] = 0.f;
  __shared__ float red[256];
  red[d] = v * v;
  __syncthreads();
  for (int s = 128; s > 0; s >>= 1) {
    if (d < s) red[d] += red[d + s];
    __syncthreads();
  }
  if (d == 0) { c2[k] = red[0]; cnt[k] = 0.f; }
}

// ---------------------------------------------------------------------------
// Per 64-row block: P = Xblk*C^T (WMMA) -> softmax -> either
//   (iterating) S += a^T * Xblk (WMMA + f32 global atomics), cnt += col sums
//   (final)     write a to global
__global__ __launch_bounds__(256) void assign_kernel(
    const float* __restrict__ X, const float* __restrict__ x2,
    const float* __restrict__ C, const float* __restrict__ c2,
    float* __restrict__ S, float* __restrict__ cnt,
    float* __restrict__ a_out) {
  extern __shared__ float smem[];
  float (*Xs)[XPAD] = (float(*)[XPAD])smem;                       // 64 x 260
  float (*Cs)[XPAD] = (float(*)[XPAD])(smem + RPB * XPAD);        // 64 x 260
  float (*Ps)[PPAD] = (float(*)[PPAD])(smem + 2 * RPB * XPAD);    // 64 x 68
  float* c2s = smem + 2 * RPB * XPAD + RPB * PPAD;                // 64

  const int tid   = threadIdx.x;
  const int lane  = tid & 31;
  const int wave  = tid >> 5;        // 0..7
  const int mlane = lane & 15;       // M / N position within tile
  const int khalf = lane >> 4;       // selects K pair {0,1} vs {2,3}
  const int row0  = blockIdx.x * RPB;

  // ---- stage X block and C into LDS via async global->LDS B128 copies ----
  {
    const int rbase = tid >> 6;        // 0..3
    const int cc    = (tid & 63) * 4;  // 0..252
    const unsigned long long xb = (unsigned long long)(uintptr_t)X;
    const unsigned long long cb = (unsigned long long)(uintptr_t)C;
#pragma unroll
    for (int i = 0; i < 16; ++i) {
      const int rr = rbase + i * 4;    // 0..63
      async_copy_b128((unsigned int)(uintptr_t)&Xs[rr][cc],
                      (unsigned int)(((row0 + rr) * DDIM + cc) * sizeof(float)),
                      xb);
      async_copy_b128((unsigned int)(uintptr_t)&Cs[rr][cc],
                      (unsigned int)((rr * DDIM + cc) * sizeof(float)), cb);
    }
  }
  if (tid < KCL) c2s[tid] = c2[tid];
  wait_asynccnt0();
  __syncthreads();

  // ---- phase 1: P(64x64) = Xblk(64x256) * C^T(256x64), v_wmma_f32_16x16x4 ----
  {
    const int rt  = wave >> 1;         // row tile 0..3
    const int ct0 = (wave & 1) * 2;    // cluster tiles ct0, ct0+1
    v8f acc0 = {0.f, 0.f, 0.f, 0.f, 0.f, 0.f, 0.f, 0.f};
    v8f acc1 = acc0;
    const float* xrow  = &Xs[rt * 16 + mlane][khalf * 2];
    const float* crow0 = &Cs[ct0 * 16 + mlane][khalf * 2];
    const float* crow1 = &Cs[(ct0 + 1) * 16 + mlane][khalf * 2];
    for (int kk = 0; kk < DDIM; kk += 4) {
      const v2f a  = *(const v2f*)(xrow + kk);   // A[m][k], m=lane, k pair by half-wave
      const v2f b0 = *(const v2f*)(crow0 + kk);  // B[k][n] = C[n][k]
      const v2f b1 = *(const v2f*)(crow1 + kk);
      acc0 = wmma_f32(a, b0, acc0);              // shared A frag, 2 indep WMMAs
      acc1 = wmma_f32(a, b1, acc1);
    }
#pragma unroll
    for (int r = 0; r < 8; ++r) {                // D: VGPR r -> M = r + 8*khalf
      Ps[rt * 16 + r + 8 * khalf][ct0 * 16 + mlane]       = acc0[r];
      Ps[rt * 16 + r + 8 * khalf][(ct0 + 1) * 16 + mlane] = acc1[r];
    }
  }
  __syncthreads();

  // ---- phase 2: logits = -2*sqrt(max(x2+c2-2P,0)); row softmax over K=64 ----
  // 4 threads per row, 16 clusters each; combine via 4-lane shuffle reductions.
  {
    const int r = tid >> 2;            // 0..63
    const int q = tid & 3;             // 0..3 -> clusters q*16..q*16+15
    const float xx = x2[row0 + r];
    float mx = -3.4e38f;
#pragma unroll 1
    for (int i = 0; i < 16; ++i) {
      const int k = q * 16 + i;
      float d2 = xx + c2s[k] - 2.0f * Ps[r][k];
      d2 = fmaxf(d2, 0.0f);
      const float lg = -2.0f * sqrtf(d2);        // -dist / TEMP, TEMP = 0.5
      Ps[r][k] = lg;
      mx = fmaxf(mx, lg);
    }
    mx = fmaxf(mx, __shfl_xor(mx, 1));
    mx = fmaxf(mx, __shfl_xor(mx, 2));
    float sum = 0.f;
#pragma unroll 1
    for (int i = 0; i < 16; ++i) {
      const int k = q * 16 + i;
      const float e = __expf(Ps[r][k] - mx);
      Ps[r][k] = e;
      sum += e;
    }
    sum += __shfl_xor(sum, 1);
    sum += __shfl_xor(sum, 2);
    const float inv = 1.0f / sum;
#pragma unroll 1
    for (int i = 0; i < 16; ++i) Ps[r][q * 16 + i] *= inv;
  }
  __syncthreads();

  if (a_out != nullptr) {                        // final pass: emit assignments
#pragma unroll
    for (int i = 0; i < 16; ++i) {
      const int idx = tid + i * 256;
      const int r = idx >> 6;
      const int k = idx & 63;
      a_out[(size_t)(row0 + r) * KCL + k] = Ps[r][k];
    }
    return;
  }

  // ---- cnt[k] += column sums of a (4 partials per cluster) ----
  {
    const int k = tid & 63;
    const int q = tid >> 6;
    float s = 0.f;
#pragma unroll
    for (int r = 0; r < 16; ++r) s += Ps[q * 16 + r][k];
    unsafeAtomicAdd(&cnt[k], s);
  }

  // ---- phase 3: S(64x256) += a^T(64x64) * Xblk(64x256), f32 WMMA ----
  {
    const int kt  = wave >> 1;          // cluster tile 0..3
    const int dt0 = (wave & 1) * 8;     // d tiles dt0..dt0+7
    v8f acc[8];
#pragma unroll
    for (int j = 0; j < 8; ++j)
      acc[j] = (v8f){0.f, 0.f, 0.f, 0.f, 0.f, 0.f, 0.f, 0.f};
    for (int rr = 0; rr < RPB; rr += 4) {
      v2f a;                             // A[m][k] = a[rr+k][cluster m]
      a.x = Ps[rr + khalf * 2][kt * 16 + mlane];
      a.y = Ps[rr + khalf * 2 + 1][kt * 16 + mlane];
#pragma unroll
      for (int j = 0; j < 8; ++j) {      // A frag reused across 8 d-tiles
        const int dc = (dt0 + j) * 16 + mlane;
        v2f b;                           // B[k][n] = X[rr+k][d n]
        b.x = Xs[rr + khalf * 2][dc];
        b.y = Xs[rr + khalf * 2 + 1][dc];
        acc[j] = wmma_f32(a, b, acc[j]);
      }
    }
#pragma unroll
    for (int j = 0; j < 8; ++j) {
      float* srow = &S[(size_t)(kt * 16 + 8 * khalf) * DDIM + (dt0 + j) * 16 + mlane];
#pragma unroll
      for (int r = 0; r < 8; ++r)        // global_atomic_add_f32 into L2
        unsafeAtomicAdd(srow + (size_t)r * DDIM, acc[j][r]);
    }
  }
}

// ---------------------------------------------------------------------------
__global__ __launch_bounds__(256) void copyC_kernel(const float* __restrict__ C,
                                                    float* __restrict__ out) {
  const int i = blockIdx.x * 256 + threadIdx.x;
  out[i] = C[i];
}

// ---------------------------------------------------------------------------
extern "C" void kernel_launch(void* const* d_in, const int* in_sizes, int n_in,
                              void* d_out, int out_size, void* d_ws, size_t ws_size,
                              hipStream_t stream) {
  const float* X      = (const float*)d_in[0];  // 16384 x 256
  const float* C_init = (const float*)d_in[1];  // 64 x 256
  float* out = (float*)d_out;                   // [C (16384) | a (1048576)]

  float* ws  = (float*)d_ws;
  float* x2  = ws;            // 16384
  float* C   = ws + 16384;    // 16384
  float* S   = ws + 32768;    // 16384
  float* c2  = ws + 49152;    // 64
  float* cnt = ws + 49216;    // 64

  x2_kernel<<<N_ROWS / 256, 256, 0, stream>>>(X, x2);
  initC_kernel<<<KCL, 256, 0, stream>>>(C_init, C, c2, S, cnt);

  for (int it = 0; it < 100; ++it) {   // fixed iteration count (reference cap)
    assign_kernel<<<NBLK, 256, SMEM_BYTES, stream>>>(X, x2, C, c2, S, cnt, nullptr);
    update_kernel<<<KCL, 256, 0, stream>>>(C, c2, S, cnt);
  }

  copyC_kernel<<<KCL, 256, 0, stream>>>(C, out);
  assign_kernel<<<NBLK, 256, SMEM_BYTES, stream>>>(X, x2, C, c2, S, cnt,
                                                   out + (size_t)KCL * DDIM);
}